// CTCLoss_62088047231154
// MI455X (gfx1250) — compile-verified
//
#include <hip/hip_runtime.h>
#include <hip/hip_bf16.h>
#include <cstdint>
#include <cstddef>

#ifndef __has_builtin
#define __has_builtin(x) 0
#endif

// Problem constants (fixed by the reference).
constexpr int T_  = 2000;
constexpr int B_  = 64;
constexpr int A_  = 32;
constexpr int L_  = 200;
constexpr int S_  = 2 * L_ + 1;   // 401 CTC states
constexpr int BLK = 416;          // 13 wave32 waves >= 401 states

// CDNA5 async global->LDS path (builtin existence proven by probe).
#if defined(__HIP_DEVICE_COMPILE__) && __has_builtin(__builtin_amdgcn_global_load_async_to_lds_b32)
#define CTC_ASYNC_LDS 1
typedef __attribute__((address_space(1))) int* ctc_as1ip;
typedef __attribute__((address_space(3))) int* ctc_as3ip;
#else
#define CTC_ASYNC_LDS 0
#endif

__device__ __forceinline__ void ctc_wait_async0() {
#if CTC_ASYNC_LDS
#if __has_builtin(__builtin_amdgcn_s_wait_asynccnt)
  __builtin_amdgcn_s_wait_asynccnt(0);
#else
  asm volatile("s_wait_asynccnt 0x0" ::: "memory");
#endif
#endif
}

// jnp.logaddexp-compatible 2-way (used only for the final combine).
__device__ __forceinline__ float lae(float a, float b) {
  float m = fmaxf(a, b);
  if (m == -__builtin_inff()) return -__builtin_inff();
  return m + log1pf(expf(-fabsf(a - b)));
}

// 3-way log-sum-exp; a0 is always finite, a1/a2 may be -inf (exp -> 0).
// Three independent v_exp_f32 pipeline; only one v_log_f32 on the chain.
__device__ __forceinline__ float lse3(float a0, float a1, float a2) {
  float m = fmaxf(fmaxf(a0, a1), a2);
  float s = expf(a0 - m) + expf(a1 - m) + expf(a2 - m);
  return m + logf(s);
}

__global__ __launch_bounds__(BLK) void ctc_alpha_kernel(
    const float* __restrict__ pred,         // [T,B,A] log-softmax
    const int*   __restrict__ target,       // [B,L]
    const int*   __restrict__ pred_lens,    // [B]
    const int*   __restrict__ target_lens,  // [B]
    float*       __restrict__ per_b)        // [B] logaddexp(f0,f1)
{
  const int b   = blockIdx.x;
  const int tid = threadIdx.x;

  __shared__ float s_alpha[2][S_ + 3];  // ping-pong alpha
  __shared__ float s_pred[2][A_];       // ping-pong log-prob row (32 floats)

  const int plen = pred_lens[b];
  const int tl   = target_lens[b];
  const int sid  = 2 * tl;
  const int tend = plen - 1;            // last timestep we need

  // Per-thread emission column: blank (0) for even states, target symbol for odd.
  const bool odd = (tid & 1) != 0;
  int sym = 0;
  if (tid < S_ && odd) sym = target[b * L_ + (tid >> 1)];
  const bool skip_ok = odd && (tid >= 3);
  const bool active  = (tid < S_);

  // t = 0 init (matches reference: all log(1e-4), states 0,1 = log(0.499)).
  if (active) {
    float v = -9.2103404f;              // logf(1e-4f)
    if (tid < 2) v = -0.69514918f;      // logf(0.499f)
    s_alpha[0][tid] = v;
  }
  // Startup: row t=1 into s_pred[1] (buffer parity == t & 1).
  if (tid < A_ && tend >= 1)
    s_pred[1][tid] = pred[((size_t)1 * B_ + b) * A_ + tid];
  __syncthreads();

  if (tend <= 0) {  // defensive; pred_lens >= T/2 in practice
    if (tid == 0) per_b[b] = lae(s_alpha[0][sid - 1], s_alpha[0][sid]);
    return;
  }

#pragma unroll 2
  for (int t = 1; t <= tend; ++t) {
    const int cur = (t - 1) & 1;
    const int nxt = t & 1;
    const float* __restrict__ ap = s_alpha[cur];
    const float* __restrict__ ep = s_pred[nxt];   // row t lives in parity t&1
    const bool   do_pref = (t < tend);

    // Warm L2 a few rows ahead (rows are 8 KB apart): global_prefetch_b8.
    if (tid == 0 && t + 8 <= tend)
      __builtin_prefetch(pred + (((size_t)(t + 8)) * B_ + b) * A_, 0, 0);

#if CTC_ASYNC_LDS
    // Kick off async DMA of row t+1 into the other pred buffer (lane-per-float).
    if (do_pref && tid < A_) {
      const float* g = pred + (((size_t)(t + 1)) * B_ + b) * A_ + tid;
      ctc_as1ip gp = (ctc_as1ip)(int*)(void*)const_cast<float*>(g);
      ctc_as3ip lp = (ctc_as3ip)(int*)(void*)&s_pred[cur][tid];
      __builtin_amdgcn_global_load_async_to_lds_b32(gp, lp, 0, 0);
    }
#else
    // Pipelined fallback: issue the load now, commit to LDS after compute.
    float prefv = 0.0f;
    if (do_pref && tid < A_)
      prefv = pred[(((size_t)(t + 1)) * B_ + b) * A_ + tid];
#endif

    if (active) {
      const float a0 = ap[tid];
      const float a1 = (tid >= 1) ? ap[tid - 1] : -__builtin_inff();
      const float a2 = skip_ok    ? ap[tid - 2] : -__builtin_inff();
      s_alpha[nxt][tid] = ep[sym] + lse3(a0, a1, a2);
    }

#if !CTC_ASYNC_LDS
    if (do_pref && tid < A_) s_pred[cur][tid] = prefv;
#endif
    ctc_wait_async0();       // async LDS writes visible before the barrier
    __syncthreads();         // one barrier per timestep
  }

  if (tid == 0) {
    const float* af = s_alpha[tend & 1];
    per_b[b] = lae(af[sid - 1], af[sid]);
  }
}

__global__ __launch_bounds__(B_) void ctc_reduce_kernel(
    const float* __restrict__ per_b, float* __restrict__ out)
{
  __shared__ float sh[B_];
  const int tid = threadIdx.x;
  sh[tid] = per_b[tid];
  __syncthreads();
  if (tid == 0) {
    float s = 0.0f;
    for (int i = 0; i < B_; ++i) s += sh[i];
    *out = -s / (float)B_;
  }
}

extern "C" void kernel_launch(void* const* d_in, const int* in_sizes, int n_in,
                              void* d_out, int out_size, void* d_ws, size_t ws_size,
                              hipStream_t stream) {
  const float* pred   = (const float*)d_in[0];  // [T,B,A] float32
  const int*   target = (const int*)  d_in[1];  // [B,L]   int32
  const int*   plens  = (const int*)  d_in[2];  // [B]
  const int*   tlens  = (const int*)  d_in[3];  // [B]
  float* per_b = (float*)d_ws;                  // 64 floats of scratch

  ctc_alpha_kernel<<<B_, BLK, 0, stream>>>(pred, target, plens, tlens, per_b);
  ctc_reduce_kernel<<<1, B_, 0, stream>>>(per_b, (float*)d_out);
  (void)in_sizes; (void)n_in; (void)out_size; (void)ws_size;
}